// FastAPLoss_88218628260841
// MI455X (gfx1250) — compile-verified
//
#include <hip/hip_runtime.h>
#include <stdint.h>

// FastAP loss, N = 4096, 10 bins (11 edges).
//
// Memory-bound: stream batch (64MB) + labels (64MB) exactly once.
// HBM floor = 128MB / 23.3 TB/s ~= 5.5us. No matmul structure (data-dependent
// soft-binning histogram) => no WMMA; the CDNA5 levers are the async
// global->LDS streaming path (ASYNCcnt) and wave32 reductions.
//
// VALU trick: accumulate CUMULATIVE histograms directly:
//   sum_{k<=l} pulse_k(t) = clamp(l+1-t, 0, 1)   (one v_med3 per bin)
// so H[l] and H_pos[l] come straight out of the accumulation (no cumsum),
// l=10 is analytically H=4096 / H_pos=N_pos, and the inner loop is
// 10 bins x 4 VALU ops per element (~43 ops per 8 streamed bytes).

#define N_ROW   4096
#define NB      10          // bins computed in the inner loop (edge 10 is free)
#define THREADS 256         // 8 waves (wave32)
#define CHUNK   1024        // elements staged per chunk (4KB per matrix)
#define NCHUNK  (N_ROW / CHUNK)

// Async copy 16 bytes/lane from global to LDS (GV addressing: per-lane 64-bit
// global address in a VGPR pair, per-lane LDS byte address in a VGPR).
// Tracked by ASYNCcnt, drained with s_wait_asynccnt.
__device__ __forceinline__ void async_b128(uint32_t lds_byte_addr, const float* gaddr) {
  asm volatile("global_load_async_to_lds_b128 %0, %1, off"
               :: "v"(lds_byte_addr), "v"(gaddr)
               : "memory");
}

__launch_bounds__(THREADS)
__global__ void fastap_rows(const float* __restrict__ batch,
                            const float* __restrict__ labels,
                            float* __restrict__ ap_out,
                            float* __restrict__ valid_out) {
  // [buf][0=batch,1=labels][elem] double-buffered stage: 16KB
  __shared__ __align__(16) float stage[2][2][CHUNK];
  __shared__ float hist[2 * NB + 1];   // [0..9]=H_pos, [10..19]=H, [20]=N_pos

  const int tid = threadIdx.x;
  const int row = blockIdx.x;
  const float* brow = batch  + (size_t)row * N_ROW;
  const float* lrow = labels + (size_t)row * N_ROW;

  if (tid < 2 * NB + 1) hist[tid] = 0.0f;

  // LDS byte address of stage[0][0][0]: low 32 bits of the flat pointer are
  // the LDS offset (aperture bits live in addr[63:32] on CDNA5).
  const uint32_t lds0 = (uint32_t)(uintptr_t)&stage[0][0][0];
  const int e0 = tid * 4;                       // 4 floats (b128) per lane

  // Prologue: issue chunk 0 into buffer 0 (ASYNCcnt -> 2 for this wave).
  async_b128(lds0 + (uint32_t)(e0 * 4),           brow + e0);
  async_b128(lds0 + (uint32_t)((CHUNK + e0) * 4), lrow + e0);

  float Hp[NB], Ha[NB];           // cumulative pos / all histograms
  float ls = 0.0f;                // labsum = N_pos contribution
#pragma unroll
  for (int l = 0; l < NB; ++l) { Hp[l] = 0.0f; Ha[l] = 0.0f; }

  for (int c = 0; c < NCHUNK; ++c) {
    const int buf = c & 1;

    if (c + 1 < NCHUNK) {
      // Prefetch next chunk into the other buffer, then wait until only the
      // prefetch pair is outstanding (async loads complete in order) so the
      // current chunk is resident.
      const int nb   = (c + 1) & 1;
      const int goff = (c + 1) * CHUNK + e0;
      async_b128(lds0 + (uint32_t)((nb * 2 * CHUNK + e0) * 4),         brow + goff);
      async_b128(lds0 + (uint32_t)((nb * 2 * CHUNK + CHUNK + e0) * 4), lrow + goff);
      asm volatile("s_wait_asynccnt 0x2" ::: "memory");
    } else {
      asm volatile("s_wait_asynccnt 0x0" ::: "memory");
    }
    __syncthreads();   // all 8 waves' async writes for chunk c have landed

    const float4 b4 = *(const float4*)&stage[buf][0][e0];  // ds_load_b128
    const float4 l4 = *(const float4*)&stage[buf][1][e0];
    const float dv[4] = {b4.x, b4.y, b4.z, b4.w};
    const float lv[4] = {l4.x, l4.y, l4.z, l4.w};

#pragma unroll
    for (int e = 0; e < 4; ++e) {
      // dist2 = 2 - 2*clip(s); t = dist2/Delta = 5 - 5*clip(s)  in [0,10]
      const float cl = __builtin_amdgcn_fmed3f(dv[e], -1.0f, 1.0f); // v_med3
      const float t  = fmaf(cl, -5.0f, 5.0f);
      const float lb = lv[e];
      ls += lb;
      // Cumulative pulse: sum_{k<=l} pulse_k(t) = clamp(l+1-t, 0, 1).
#pragma unroll
      for (int l = 0; l < NB; ++l) {
        const float u = (float)(l + 1) - t;                      // v_sub
        const float cp = __builtin_amdgcn_fmed3f(u, 0.0f, 1.0f); // v_med3
        Hp[l] = fmaf(cp, lb, Hp[l]);                             // v_fma
        Ha[l] += cp;                                             // v_add
      }
    }
    __syncthreads();   // guard: buffer 'buf' is overwritten 2 chunks later
  }

  // Wave32 butterfly reduction, then one ds_add_f32 per wave per slot.
#pragma unroll
  for (int l = 0; l < NB; ++l) {
    float vp = Hp[l], va = Ha[l];
#pragma unroll
    for (int off = 16; off >= 1; off >>= 1) {
      vp += __shfl_xor(vp, off, 32);
      va += __shfl_xor(va, off, 32);
    }
    if ((tid & 31) == 0) {
      atomicAdd(&hist[l],      vp);
      atomicAdd(&hist[NB + l], va);
    }
  }
  {
    float vs = ls;
#pragma unroll
    for (int off = 16; off >= 1; off >>= 1) vs += __shfl_xor(vs, off, 32);
    if ((tid & 31) == 0) atomicAdd(&hist[2 * NB], vs);
  }
  __syncthreads();

  if (tid == 0) {
    // ap = sum_l h_pos[l] * H_pos[l] / H[l],  h_pos[l] = H_pos[l]-H_pos[l-1].
    const float Np = hist[2 * NB];           // H_pos[10] = N_pos
    float prev = 0.0f, ap = 0.0f;
#pragma unroll
    for (int l = 0; l < NB; ++l) {
      const float hpl = hist[l];             // H_pos[l]
      const float hal = hist[NB + l];        // H[l]
      const float dh  = hpl - prev;
      prev = hpl;
      if (hal > 0.0f) ap += dh * hpl / hal;
    }
    // Last edge: H[10] = N_ROW exactly (pulses partition unity), H_pos[10]=Np.
    ap += (Np - prev) * Np / (float)N_ROW;
    ap_out[row]    = (Np > 0.0f) ? (ap / Np) : 0.0f;
    valid_out[row] = (Np > 0.0f) ? 1.0f : 0.0f;
  }
}

__launch_bounds__(THREADS)
__global__ void fastap_final(const float* __restrict__ ap,
                             const float* __restrict__ valid,
                             float* __restrict__ out) {
  __shared__ float sa[8], sv[8];
  float a = 0.0f, v = 0.0f;
  for (int i = threadIdx.x; i < N_ROW; i += THREADS) {
    a += ap[i];
    v += valid[i];
  }
#pragma unroll
  for (int off = 16; off >= 1; off >>= 1) {
    a += __shfl_xor(a, off, 32);
    v += __shfl_xor(v, off, 32);
  }
  const int wave = threadIdx.x >> 5;
  const int lane = threadIdx.x & 31;
  if (lane == 0) { sa[wave] = a; sv[wave] = v; }
  __syncthreads();
  if (threadIdx.x == 0) {
    float ta = 0.0f, tv = 0.0f;
#pragma unroll
    for (int w = 0; w < 8; ++w) { ta += sa[w]; tv += sv[w]; }
    out[0] = 1.0f - ta / tv;
  }
}

extern "C" void kernel_launch(void* const* d_in, const int* in_sizes, int n_in,
                              void* d_out, int out_size, void* d_ws, size_t ws_size,
                              hipStream_t stream) {
  const float* batch  = (const float*)d_in[0];
  const float* labels = (const float*)d_in[1];
  float* out = (float*)d_out;

  float* ws_ap    = (float*)d_ws;            // [4096]
  float* ws_valid = ws_ap + N_ROW;           // [4096]

  fastap_rows<<<N_ROW, THREADS, 0, stream>>>(batch, labels, ws_ap, ws_valid);
  fastap_final<<<1, THREADS, 0, stream>>>(ws_ap, ws_valid, out);
}